// CSV_20727512170902
// MI455X (gfx1250) — compile-verified
//
#include <hip/hip_runtime.h>

#define VOCAB 100000
#define SIZE 128
#define TWO_W 10
#define NEG 10
#define BATCH 8192
#define ROWS_PER_BLOCK 16
#define LSTRIDE 132   // padded LDS row stride (floats): 528B = 33*16 -> 16B aligned, conflict-free

typedef __attribute__((ext_vector_type(2))) float v2f;
typedef __attribute__((ext_vector_type(8))) float v8f;

// -log_sigmoid(y) = log(1 + exp(-y)); y is pre-clipped to [-10, 10]
__device__ __forceinline__ float neg_log_sigmoid(float y) {
    return log1pf(__expf(-y));
}

// Issue 16 rows of sense-embedding gather directly into LDS via the CDNA5
// async global->LDS path (ASYNCcnt-tracked, no VGPR round-trip).
__device__ __forceinline__ void issue_sense_gather(const float* __restrict__ sense_embs,
                                                   const int* __restrict__ data,
                                                   int row0, int slot,
                                                   float* buf, int lane)
{
    const int col = (slot < NEG) ? (12 + slot) : 11;  // neg cols 12..21, pos col 11
    #pragma unroll
    for (int r = 0; r < ROWS_PER_BLOCK; ++r) {
        const int sidx = data[(size_t)(row0 + r) * 32 + col];  // uniform -> scalarized
        const unsigned long long gaddr =
            (unsigned long long)(uintptr_t)(sense_embs + (size_t)sidx * SIZE + lane * 4);
        // Low 32 bits of a generic shared pointer == LDS byte offset
        // (shared aperture base lives in addr[63:32] per the ISA aperture check).
        const unsigned int ldso =
            (unsigned int)(uintptr_t)(buf + r * LSTRIDE + lane * 4);
        asm volatile("global_load_async_to_lds_b128 %0, %1, off"
                     :: "v"(ldso), "v"(gaddr)
                     : "memory");
    }
}

__global__ void __launch_bounds__(32)
w2v_sg_loss_kernel(const int* __restrict__ data,
                   const float* __restrict__ global_embs,
                   const float* __restrict__ sense_embs,
                   const float* __restrict__ ctx_weight,
                   float* __restrict__ out)
{
    __shared__ alignas(16) float featbuf[16 * LSTRIDE];       // ctx_feats 16x128 (padded)
    __shared__ alignas(16) float sensebuf[2][16 * LSTRIDE];   // double-buffered sense tiles
    __shared__ float ipbuf[16];                               // per-row inner products

    const int lane = threadIdx.x;            // wave32
    const int row0 = blockIdx.x * ROWS_PER_BLOCK;
    const int half = lane >> 4;              // 0: lanes 0-15, 1: lanes 16-31
    const int rsel = lane & 15;              // WMMA M (A) / N (B) index for this lane

    // Kick off slot 0's sense gather immediately; it streams into LDS while
    // phase 1 (ctx features) runs.
    issue_sense_gather(sense_embs, data, row0, 0, sensebuf[0], lane);

    // Preload ctx_weight for this lane's 4 dims (10 x float4, L2-hot)
    float4 wvec[TWO_W];
    #pragma unroll
    for (int w = 0; w < TWO_W; ++w)
        wvec[w] = *(const float4*)(ctx_weight + w * SIZE + lane * 4);

    // ---- Phase 1: gather + weight-sum 10 ctx embeddings per row -> featbuf ----
    for (int r = 0; r < ROWS_PER_BLOCK; ++r) {
        const int* drow = data + (size_t)(row0 + r) * 32;
        float4 f = make_float4(0.f, 0.f, 0.f, 0.f);
        #pragma unroll
        for (int w = 0; w < TWO_W; ++w) {
            const int idx = drow[w];  // uniform across lanes -> scalarized
            float4 e = *(const float4*)(global_embs + (size_t)idx * SIZE + lane * 4);
            f.x = fmaf(e.x, wvec[w].x, f.x);
            f.y = fmaf(e.y, wvec[w].y, f.y);
            f.z = fmaf(e.z, wvec[w].z, f.z);
            f.w = fmaf(e.w, wvec[w].w, f.w);
        }
        *(float4*)(&featbuf[r * LSTRIDE + lane * 4]) = f;
    }

    float acc = 0.0f;

    // ---- Phase 2: 11 slots (10 neg + 1 pos); double-buffered async gathers
    //      overlapped with a 16x16x128 f32 WMMA block per slot ----
    for (int slot = 0; slot < NEG + 1; ++slot) {
        const float* cur = sensebuf[slot & 1];

        if (slot < NEG) {
            // Prefetch next slot into the other buffer, then wait until only
            // those 16 remain outstanding => current slot's data is in LDS.
            issue_sense_gather(sense_embs, data, row0, slot + 1,
                               sensebuf[(slot + 1) & 1], lane);
            asm volatile("s_wait_asynccnt 0x10" ::: "memory");
        } else {
            asm volatile("s_wait_asynccnt 0x0" ::: "memory");
        }
        __syncthreads();  // orders featbuf stores (first iter) & ipbuf reuse

        // C(16x16,f32) = A(16x128) x B(128x16) via 32 chained v_wmma_f32_16x16x4_f32
        // A fragment: lane holds (M=rsel, K=k0+2*half, +1); B symmetric with N=rsel.
        v8f c = {0.f, 0.f, 0.f, 0.f, 0.f, 0.f, 0.f, 0.f};
        const float* fb = &featbuf[rsel * LSTRIDE + half * 2];
        const float* sb = &cur[rsel * LSTRIDE + half * 2];
        #pragma unroll
        for (int k0 = 0; k0 < SIZE; k0 += 4) {
            v2f a, b;
            a.x = fb[k0]; a.y = fb[k0 + 1];
            b.x = sb[k0]; b.y = sb[k0 + 1];
            c = __builtin_amdgcn_wmma_f32_16x16x4_f32(
                    /*neg_a=*/false, a, /*neg_b=*/false, b,
                    /*c_mod=*/(short)0, c, /*reuse_a=*/false, /*reuse_b=*/false);
        }

        // Extract diagonal C[m][m]: VGPR j holds (M=j+8*half, N=rsel); diag when M==N.
        const int wantJ = rsel - half * 8;
        float diag = 0.0f;
        #pragma unroll
        for (int j = 0; j < 8; ++j)
            diag = (wantJ == j) ? c[j] : diag;
        if (wantJ >= 0 && wantJ < 8)
            ipbuf[rsel] = diag;
        __syncthreads();

        // Loss on lanes 0..15 (lane == batch row within tile)
        if (lane < 16) {
            const float ip = ipbuf[lane];
            if (slot < NEG) {
                const float m = (float)data[(size_t)(row0 + lane) * 32 + 22 + slot];
                const float y = fminf(fmaxf(-ip, -10.f), 10.f);
                acc = fmaf(neg_log_sigmoid(y), m, acc);
            } else {
                const float y = fminf(fmaxf(ip, -10.f), 10.f);
                acc += neg_log_sigmoid(y);
            }
        }
    }

    // Wave reduce (lanes 16..31 hold 0) then one atomic per block
    #pragma unroll
    for (int m = 16; m >= 1; m >>= 1)
        acc += __shfl_xor(acc, m, 32);
    if (lane == 0)
        atomicAdd(out, acc);
}

__global__ void zero_out_kernel(float* o) {
    o[0] = 0.0f;
}

extern "C" void kernel_launch(void* const* d_in, const int* in_sizes, int n_in,
                              void* d_out, int out_size, void* d_ws, size_t ws_size,
                              hipStream_t stream) {
    const int*   data        = (const int*)d_in[0];
    const float* global_embs = (const float*)d_in[1];
    const float* sense_embs  = (const float*)d_in[2];
    const float* ctx_weight  = (const float*)d_in[3];
    float* out = (float*)d_out;

    zero_out_kernel<<<1, 1, 0, stream>>>(out);
    w2v_sg_loss_kernel<<<BATCH / ROWS_PER_BLOCK, 32, 0, stream>>>(
        data, global_embs, sense_embs, ctx_weight, out);
}